// FourierLayer_73040213836314
// MI455X (gfx1250) — compile-verified
//
#include <hip/hip_runtime.h>
#include <math.h>

// ---------------------------------------------------------------------------
// FNO spectral layer for MI455X (gfx1250, wave32).
// Partial-DFT formulation: everything becomes 16x16x4 fp32 WMMA GEMMs.
//   B=16, H=W=128, Cin=Cout=64, MODES=16.
// Scratch layout (floats):
//   Tr/Ti : [b][kw][h][i]      16*16*128*64 = 2,097,152 each  (8 MiB each)
//   Xr/Xi : [kh][kw][i][b]     16*16*64*16  =   262,144 each  (1 MiB each)
//   Wr/Wi : [kh][kw][i][o]     16*16*64*64  = 1,048,576 each  (4 MiB each)
//   Cr/Ci : [b][kh][kw][o]     16*16*16*64  =   262,144 each  (1 MiB each)
// Total ws = 28 MiB.
// ---------------------------------------------------------------------------

typedef __attribute__((ext_vector_type(2))) float v2f;
typedef __attribute__((ext_vector_type(8))) float v8f;

__device__ __forceinline__ v8f wmma4(v2f a, v2f b, v8f c) {
  // D = A(16x4 f32) * B(4x16 f32) + C(16x16 f32)
  return __builtin_amdgcn_wmma_f32_16x16x4_f32(false, a, false, b, (short)0, c,
                                               false, false);
}

#define PI_OVER_64 0.049087385212340519f  // 2*pi/128

// ---------------------------------------------------------------------------
// Kernel 1: partial DFT over w.  Block = one (b,h) row, 4 waves.
// T[kw][i] = sum_w x[w][i] * e^{-2pi i kw w/128}   (Tr = sum x*cos, Ti = sum x*(-sin))
// GEMM: M=kw(16), N=i (4 tiles of 16, one per wave), K=w(128).
// ---------------------------------------------------------------------------
__global__ __launch_bounds__(128) void k_dftw(const float* __restrict__ x,
                                              float* __restrict__ Tr,
                                              float* __restrict__ Ti) {
  __shared__ float xrow[128 * 64];    // [w][i]
  __shared__ float bw[128][16][2];    // [w][kw] {cos, -sin}
  const int bh = blockIdx.x;          // b*128 + h
  const int b = bh >> 7, h = bh & 127;
  const int tid = threadIdx.x;

  const float* xp = x + (size_t)bh * (128 * 64);
  for (int idx = tid; idx < 128 * 64; idx += 128) xrow[idx] = xp[idx];
  for (int e = tid; e < 128 * 16; e += 128) {
    int w = e >> 4, kw = e & 15;
    float s, c;
    __sincosf((float)(w * kw) * PI_OVER_64, &s, &c);
    bw[w][kw][0] = c;
    bw[w][kw][1] = -s;
  }
  __syncthreads();

  const int lane = tid & 31, wave = tid >> 5;
  const int i0 = wave * 16;
  const int mn = lane & 15;            // m (=kw) for A, n (=i) for B/D
  const int k2 = (lane >> 4) * 2;      // k-pair select per lane half

  v8f ar = {}, ai = {};
  for (int kb = 0; kb < 128; kb += 4) {
    int k0 = kb + k2;
    v2f ac, as, bx;
    ac.x = bw[k0][mn][0];      ac.y = bw[k0 + 1][mn][0];
    as.x = bw[k0][mn][1];      as.y = bw[k0 + 1][mn][1];
    bx.x = xrow[k0 * 64 + i0 + mn];
    bx.y = xrow[(k0 + 1) * 64 + i0 + mn];
    ar = wmma4(ac, bx, ar);
    ai = wmma4(as, bx, ai);
  }
  for (int r = 0; r < 8; ++r) {
    int kw = r + ((lane >> 4) << 3);
    size_t a = (((size_t)(b * 16 + kw)) * 128 + h) * 64 + i0 + mn;
    Tr[a] = ar[r];
    Ti[a] = ai[r];
  }
}

// ---------------------------------------------------------------------------
// Kernel 2: partial DFT over h.  Block = one (b,kw), 4 waves.
// X[kh][i] = sum_h e^{-i th}/128 * T[h][i]  (complex*complex)
//   Xr = sum c*Tr + s*Ti ; Xi = sum c*Ti - s*Tr
// GEMM: M=kh(16), N=i(4 tiles), K=h(128).
// ---------------------------------------------------------------------------
__global__ __launch_bounds__(128) void k_dfth(const float* __restrict__ Tr,
                                              const float* __restrict__ Ti,
                                              float* __restrict__ Xr,
                                              float* __restrict__ Xi) {
  __shared__ float bh[128][16][2];  // [h][kh] {cos/128, sin/128}
  const int blk = blockIdx.x;       // b*16 + kw
  const int b = blk >> 4, kw = blk & 15;
  const int tid = threadIdx.x;
  for (int e = tid; e < 128 * 16; e += 128) {
    int h = e >> 4, kh = e & 15;
    float s, c;
    __sincosf((float)(h * kh) * PI_OVER_64, &s, &c);
    bh[h][kh][0] = c * (1.0f / 128.0f);
    bh[h][kh][1] = s * (1.0f / 128.0f);
  }
  __syncthreads();

  const int lane = tid & 31, wave = tid >> 5;
  const int i0 = wave * 16;
  const int mn = lane & 15;
  const int k2 = (lane >> 4) * 2;
  const float* trp = Tr + (size_t)blk * (128 * 64);  // [h][i]
  const float* tip = Ti + (size_t)blk * (128 * 64);

  v8f aR = {}, aI1 = {}, aI2 = {};
  for (int kb = 0; kb < 128; kb += 4) {
    int k0 = kb + k2;
    v2f ac, as, btr, bti;
    ac.x = bh[k0][mn][0];      ac.y = bh[k0 + 1][mn][0];
    as.x = bh[k0][mn][1];      as.y = bh[k0 + 1][mn][1];
    btr.x = trp[k0 * 64 + i0 + mn]; btr.y = trp[(k0 + 1) * 64 + i0 + mn];
    bti.x = tip[k0 * 64 + i0 + mn]; bti.y = tip[(k0 + 1) * 64 + i0 + mn];
    aR = wmma4(ac, btr, aR);
    aR = wmma4(as, bti, aR);
    aI1 = wmma4(ac, bti, aI1);
    aI2 = wmma4(as, btr, aI2);
  }
  // store X as [kh][kw][i][b] (A-matrix-friendly for the mode-mix GEMM)
  for (int r = 0; r < 8; ++r) {
    int kh = r + ((lane >> 4) << 3);
    size_t a = (((size_t)kh * 16 + kw) * 64 + (i0 + mn)) * 16 + b;
    Xr[a] = aR[r];
    Xi[a] = aI1[r] - aI2[r];
  }
}

// ---------------------------------------------------------------------------
// Kernel 3: repack weights (i,o,kh,kw) -> (kh,kw,i,o) for coalesced B tiles.
// ---------------------------------------------------------------------------
__global__ void k_repack(const float* __restrict__ wr,
                         const float* __restrict__ wi,
                         float* __restrict__ Wr, float* __restrict__ Wi) {
  int idx = blockIdx.x * blockDim.x + threadIdx.x;  // over 16*16*64*64
  if (idx >= 16 * 16 * 64 * 64) return;
  int o = idx & 63, i = (idx >> 6) & 63, kw = (idx >> 12) & 15, kh = idx >> 16;
  size_t src = (((size_t)i * 64 + o) * 16 + kh) * 16 + kw;
  Wr[idx] = wr[src];
  Wi[idx] = wi[src];
}

// ---------------------------------------------------------------------------
// Kernel 4: per-mode complex channel mix.  Block = one (kh,kw), 4 waves.
// out[b][o] = sum_i X[b][i] * W[i][o]  (complex): M=b(16), K=i(64), N=o(4 tiles)
// ---------------------------------------------------------------------------
__global__ __launch_bounds__(128) void k_mix(const float* __restrict__ Xr,
                                             const float* __restrict__ Xi,
                                             const float* __restrict__ Wr,
                                             const float* __restrict__ Wi,
                                             float* __restrict__ Cr,
                                             float* __restrict__ Ci) {
  const int mode = blockIdx.x;  // kh*16 + kw
  const int kh = mode >> 4, kw = mode & 15;
  const int tid = threadIdx.x, lane = tid & 31, wave = tid >> 5;
  const int o0 = wave * 16;
  const int mn = lane & 15;
  const int k2 = (lane >> 4) * 2;
  const float* xr = Xr + (size_t)mode * (64 * 16);   // [i][b]
  const float* xi = Xi + (size_t)mode * (64 * 16);
  const float* wrp = Wr + (size_t)mode * (64 * 64);  // [i][o]
  const float* wip = Wi + (size_t)mode * (64 * 64);

  v8f aRR = {}, aII = {}, aRI = {}, aIR = {};
  for (int kb = 0; kb < 64; kb += 4) {
    int k0 = kb + k2;
    v2f axr, axi, bwr, bwi;
    axr.x = xr[k0 * 16 + mn];        axr.y = xr[(k0 + 1) * 16 + mn];
    axi.x = xi[k0 * 16 + mn];        axi.y = xi[(k0 + 1) * 16 + mn];
    bwr.x = wrp[k0 * 64 + o0 + mn];  bwr.y = wrp[(k0 + 1) * 64 + o0 + mn];
    bwi.x = wip[k0 * 64 + o0 + mn];  bwi.y = wip[(k0 + 1) * 64 + o0 + mn];
    aRR = wmma4(axr, bwr, aRR);
    aII = wmma4(axi, bwi, aII);
    aRI = wmma4(axr, bwi, aRI);
    aIR = wmma4(axi, bwr, aIR);
  }
  // store C' as [b][kh][kw][o]
  for (int r = 0; r < 8; ++r) {
    int bb = r + ((lane >> 4) << 3);
    size_t a = (((size_t)bb * 16 + kh) * 16 + kw) * 64 + o0 + mn;
    Cr[a] = aRR[r] - aII[r];
    Ci[a] = aRI[r] + aIR[r];
  }
}

// ---------------------------------------------------------------------------
// Kernel 5: fused inverse partial DFT.
// Block = (b, h-tile of 16, o-tile of 16), 4 waves.
// Phase 1: D[h][kw][o] = sum_kh e^{+i th} * C'[kh][kw][o]  -> LDS
// Phase 2: y[h][w][o]  = sum_kw (cos*s(kw)/128)*Dr - (sin*s(kw)/128)*Di
//   (kw=0 term uses s=1 and drops Di -> matches irfft dropping Im of bin 0)
// ---------------------------------------------------------------------------
__global__ __launch_bounds__(128) void k_idft(const float* __restrict__ Cr,
                                              const float* __restrict__ Ci,
                                              float* __restrict__ y) {
  const int blk = blockIdx.x;  // b*32 + ht*4 + ot
  const int b = blk >> 5, ht = (blk >> 2) & 7, ot = blk & 3;
  const int h0 = ht * 16, o0 = ot * 16;
  __shared__ float bhs[16][16][2];   // [h-in-tile][kh] {cos, sin}
  __shared__ float bws[128][16][2];  // [w][kw] {cos*s/128, -sin*s/128}
  __shared__ float Dr[16][16][16];   // [h][kw][o]
  __shared__ float Di[16][16][16];
  const int tid = threadIdx.x, lane = tid & 31, wave = tid >> 5;
  const int mn = lane & 15;
  const int k2 = (lane >> 4) * 2;

  for (int e = tid; e < 16 * 16; e += 128) {
    int hh = e >> 4, kh = e & 15;
    float s, c;
    __sincosf((float)((h0 + hh) * kh) * PI_OVER_64, &s, &c);
    bhs[hh][kh][0] = c;
    bhs[hh][kh][1] = s;
  }
  for (int e = tid; e < 128 * 16; e += 128) {
    int w = e >> 4, kw = e & 15;
    float s, c;
    __sincosf((float)(w * kw) * PI_OVER_64, &s, &c);
    float sc = (kw == 0 ? 1.0f : 2.0f) * (1.0f / 128.0f);
    bws[w][kw][0] = c * sc;
    bws[w][kw][1] = -s * sc;
  }
  __syncthreads();

  // Phase 1: each wave handles 4 kw values. M=h(16), N=o(16), K=kh(16).
  for (int kwq = 0; kwq < 4; ++kwq) {
    int kw = wave * 4 + kwq;
    v8f aR1 = {}, aR2 = {}, aI = {};
    for (int kb = 0; kb < 16; kb += 4) {
      int k0 = kb + k2;
      v2f ac, as, bcr, bci;
      ac.x = bhs[mn][k0][0];  ac.y = bhs[mn][k0 + 1][0];
      as.x = bhs[mn][k0][1];  as.y = bhs[mn][k0 + 1][1];
      size_t base = (((size_t)b * 16 + k0) * 16 + kw) * 64 + o0 + mn;
      bcr.x = Cr[base];  bcr.y = Cr[base + 16 * 64];  // kh stride = 1024
      bci.x = Ci[base];  bci.y = Ci[base + 16 * 64];
      // Dr = sum c*Cr - s*Ci ; Di = sum c*Ci + s*Cr
      aR1 = wmma4(ac, bcr, aR1);
      aR2 = wmma4(as, bci, aR2);
      aI = wmma4(ac, bci, aI);
      aI = wmma4(as, bcr, aI);
    }
    for (int r = 0; r < 8; ++r) {
      int hh = r + ((lane >> 4) << 3);
      Dr[hh][kw][mn] = aR1[r] - aR2[r];
      Di[hh][kw][mn] = aI[r];
    }
  }
  __syncthreads();

  // Phase 2: units (h, w-tile): M=w(16), N=o(16), K=kw(16). Single accumulator
  // because the -sin is folded into the basis.
  for (int u = wave; u < 16 * 8; u += 4) {
    int hh = u >> 3, w0 = (u & 7) * 16;
    v8f acc = {};
    for (int kb = 0; kb < 16; kb += 4) {
      int k0 = kb + k2;
      v2f ac, ans, bdr, bdi;
      ac.x = bws[w0 + mn][k0][0];   ac.y = bws[w0 + mn][k0 + 1][0];
      ans.x = bws[w0 + mn][k0][1];  ans.y = bws[w0 + mn][k0 + 1][1];
      bdr.x = Dr[hh][k0][mn];       bdr.y = Dr[hh][k0 + 1][mn];
      bdi.x = Di[hh][k0][mn];       bdi.y = Di[hh][k0 + 1][mn];
      acc = wmma4(ac, bdr, acc);
      acc = wmma4(ans, bdi, acc);
    }
    for (int r = 0; r < 8; ++r) {
      int ww = w0 + r + ((lane >> 4) << 3);
      size_t a = (((size_t)(b * 128 + h0 + hh)) * 128 + ww) * 64 + o0 + mn;
      y[a] = acc[r];  // lanes run over o -> 64B contiguous per half-wave
    }
  }
}

// ---------------------------------------------------------------------------
extern "C" void kernel_launch(void* const* d_in, const int* in_sizes, int n_in,
                              void* d_out, int out_size, void* d_ws,
                              size_t ws_size, hipStream_t stream) {
  const float* x = (const float*)d_in[0];   // (16,128,128,64)
  const float* wr = (const float*)d_in[1];  // (64,64,16,16)
  const float* wi = (const float*)d_in[2];
  float* y = (float*)d_out;                 // (16,128,128,64)

  char* ws = (char*)d_ws;
  const size_t nT = (size_t)16 * 16 * 128 * 64;  // 2,097,152
  const size_t nX = (size_t)16 * 16 * 64 * 16;   //   262,144
  const size_t nW = (size_t)16 * 16 * 64 * 64;   // 1,048,576
  const size_t nC = (size_t)16 * 16 * 16 * 64;   //   262,144
  float* Tr = (float*)ws;                  ws += nT * 4;
  float* Ti = (float*)ws;                  ws += nT * 4;
  float* Xr = (float*)ws;                  ws += nX * 4;
  float* Xi = (float*)ws;                  ws += nX * 4;
  float* Wr = (float*)ws;                  ws += nW * 4;
  float* Wi = (float*)ws;                  ws += nW * 4;
  float* Crp = (float*)ws;                 ws += nC * 4;
  float* Cip = (float*)ws;                 // total 28 MiB

  k_dftw<<<16 * 128, 128, 0, stream>>>(x, Tr, Ti);
  k_repack<<<(int)(nW / 256), 256, 0, stream>>>(wr, wi, Wr, Wi);
  k_dfth<<<16 * 16, 128, 0, stream>>>(Tr, Ti, Xr, Xi);
  k_mix<<<256, 128, 0, stream>>>(Xr, Xi, Wr, Wi, Crp, Cip);
  k_idft<<<16 * 8 * 4, 128, 0, stream>>>(Crp, Cip, y);
}